// ThreeNN_15006615733861
// MI455X (gfx1250) — compile-verified
//
#include <hip/hip_runtime.h>

typedef float v2f __attribute__((ext_vector_type(2)));
typedef float v8f __attribute__((ext_vector_type(8)));

#define B_  8
#define NQ  16384
#define NM  4096
#define QT  16          // queries per wave (one WMMA A tile)
#define WAVES_PER_BLOCK 8

// Insert candidate (d, j) into sorted top-3 (d0<=d1<=d2) + matching indices.
// Pure compare + single-level selects (each lowers to v_cndmask sharing the
// compare's VCC; NO nested ternaries -> no exec-mask branches, and no
// fmin/fmax canonicalization ops).
__device__ __forceinline__ void insert3(float d, int j,
                                        float& d0, float& d1, float& d2,
                                        int& i0, int& i1, int& i2) {
    const bool c0 = d < d0;                  // stage 0: (d,j) vs (d0,i0)
    const float lo0 = c0 ? d  : d0;
    const float hi0 = c0 ? d0 : d;
    const int   jl0 = c0 ? j  : i0;
    const int   jh0 = c0 ? i0 : j;

    const bool c1 = hi0 < d1;                // stage 1: loser vs (d1,i1)
    const float lo1 = c1 ? hi0 : d1;
    const float hi1 = c1 ? d1  : hi0;
    const int   jl1 = c1 ? jh0 : i1;
    const int   jh1 = c1 ? i1  : jh0;

    const bool c2 = hi1 < d2;                // stage 2: loser vs (d2,i2)
    d0 = lo0;  i0 = jl0;
    d1 = lo1;  i1 = jl1;
    d2 = c2 ? hi1 : d2;
    i2 = c2 ? jh1 : i2;
}

__global__ void __launch_bounds__(256)
three_nn_wmma(const float* __restrict__ unknown,   // [B, NQ, 3]
              const float* __restrict__ known,     // [B, NM, 3]
              float* __restrict__ outDist,         // [B, NQ, 3]
              int*   __restrict__ outIdx) {        // [B, NQ, 3]
    const int lane = threadIdx.x & 31;             // wave32
    const int wv   = blockIdx.x * WAVES_PER_BLOCK + (threadIdx.x >> 5);
    const int tilesPerBatch = NQ / QT;             // 1024
    const int b  = wv / tilesPerBatch;
    const int q0 = (wv % tilesPerBatch) * QT;

    const int  col = lane & 15;                    // ref column / query row for A
    const bool hi  = lane >= 16;                   // upper half: K=2,3 for A/B

    // ---- A tile: row m = [-2qx, -2qy, -2qz, 1] (16x4 f32, 2 VGPRs) ----
    const float* qp = unknown + ((size_t)b * NQ + q0 + col) * 3;
    v2f A;
    A.x = -2.0f * (hi ? qp[2] : qp[0]);
    A.y =  hi ? 1.0f : (-2.0f * qp[1]);

    // ---- per-lane top-3 state for 8 query slots ----
    float d0[8], d1[8], d2[8];
    int   i0[8], i1[8], i2[8];
#pragma unroll
    for (int s = 0; s < 8; ++s) {
        d0[s] = d1[s] = d2[s] = 3.402823466e38f;
        i0[s] = i1[s] = i2[s] = 0;
    }

    const float* kb = known + (size_t)b * NM * 3;

    // ---- scan refs in 16-wide tiles: WMMA computes |p|^2 - 2 q.p for 16x16 pairs ----
#pragma unroll 2
    for (int t = 0; t < NM / 16; ++t) {
        const int j = t * 16 + col;                // this lane's ref column
        const float kx = kb[j * 3 + 0];
        const float ky = kb[j * 3 + 1];
        const float kz = kb[j * 3 + 2];
        const float p2 = __builtin_fmaf(kz, kz, __builtin_fmaf(ky, ky, kx * kx));

        // B tile: col n = [px, py, pz, p2] (4x16 f32, 2 VGPRs)
        v2f Bm;
        Bm.x = hi ? kz : kx;
        Bm.y = hi ? p2 : ky;

        v8f C = {0.f, 0.f, 0.f, 0.f, 0.f, 0.f, 0.f, 0.f};
        // D[m, n] = p2[n] - 2 * dot(q[m], p[n])   (16x16 f32 tile, 8 VGPRs)
        v8f D = __builtin_amdgcn_wmma_f32_16x16x4_f32(
            false, A, false, Bm, (short)0, C, false, false);

#pragma unroll
        for (int s = 0; s < 8; ++s)
            insert3(D[s], j, d0[s], d1[s], d2[s], i0[s], i1[s], i2[s]);
    }

    // ---- butterfly merge across the 16 lanes sharing each query half ----
    // masks 1,2,4,8 stay inside each 16-lane half (lanes 0-15 hold queries
    // q0..q0+7 in slots 0..7; lanes 16-31 hold q0+8..q0+15).
#pragma unroll
    for (int m = 1; m <= 8; m <<= 1) {
#pragma unroll
        for (int s = 0; s < 8; ++s) {
            float e0 = __shfl_xor(d0[s], m, 32);
            float e1 = __shfl_xor(d1[s], m, 32);
            float e2 = __shfl_xor(d2[s], m, 32);
            int   j0 = __shfl_xor(i0[s], m, 32);
            int   j1 = __shfl_xor(i1[s], m, 32);
            int   j2 = __shfl_xor(i2[s], m, 32);
            insert3(e0, j0, d0[s], d1[s], d2[s], i0[s], i1[s], i2[s]);
            insert3(e1, j1, d0[s], d1[s], d2[s], i0[s], i1[s], i2[s]);
            insert3(e2, j2, d0[s], d1[s], d2[s], i0[s], i1[s], i2[s]);
        }
    }

    // ---- writeout: lane 0 -> queries q0..q0+7, lane 16 -> q0+8..q0+15 ----
    if (col == 0) {
        const int qb = q0 + (hi ? 8 : 0);
#pragma unroll
        for (int s = 0; s < 8; ++s) {
            const int q = qb + s;
            const float* up = unknown + ((size_t)b * NQ + q) * 3;
            const float ux = up[0], uy = up[1], uz = up[2];
            const float q2 = __builtin_fmaf(uz, uz, __builtin_fmaf(uy, uy, ux * ux));
            const size_t o = ((size_t)b * NQ + q) * 3;
            outDist[o + 0] = d0[s] + q2;
            outDist[o + 1] = d1[s] + q2;
            outDist[o + 2] = d2[s] + q2;
            outIdx[o + 0] = i0[s];
            outIdx[o + 1] = i1[s];
            outIdx[o + 2] = i2[s];
        }
    }
}

extern "C" void kernel_launch(void* const* d_in, const int* in_sizes, int n_in,
                              void* d_out, int out_size, void* d_ws, size_t ws_size,
                              hipStream_t stream) {
    (void)in_sizes; (void)n_in; (void)d_ws; (void)ws_size; (void)out_size;
    const float* unknown = (const float*)d_in[0];
    const float* known   = (const float*)d_in[1];
    float* outDist = (float*)d_out;
    int*   outIdx  = (int*)d_out + (size_t)B_ * NQ * 3;   // second tuple element

    const int totalWaves = (B_ * NQ) / QT;                // 8192
    const int blocks = totalWaves / WAVES_PER_BLOCK;      // 1024 blocks x 256 thr
    three_nn_wmma<<<blocks, 256, 0, stream>>>(unknown, known, outDist, outIdx);
}